// MeshMeshIntersection_87875030876608
// MI455X (gfx1250) — compile-verified
//
#include <hip/hip_runtime.h>
#include <hip/hip_bf16.h>
#include <stdint.h>

#define MAXC 32
#define EPS_T 1e-8f
#define CHUNK 256                 // target triangles per LDS chunk (9216 B)
#define BLOCK_THREADS 256
#define WAVES_PER_BLOCK (BLOCK_THREADS / 32)

typedef int v4i_t __attribute__((ext_vector_type(4)));
typedef __attribute__((address_space(1))) v4i_t* g_v4i_ptr;
typedef __attribute__((address_space(3))) v4i_t* l_v4i_ptr;

// ---------------- CDNA5 async global->LDS helpers ----------------
__device__ __forceinline__ void async_copy_b128(float4* lds_dst, const float4* gsrc) {
#if __has_builtin(__builtin_amdgcn_global_load_async_to_lds_b128)
  __builtin_amdgcn_global_load_async_to_lds_b128(
      (g_v4i_ptr)(v4i_t*)(float4*)gsrc,   // global src (AS1, const stripped)
      (l_v4i_ptr)(v4i_t*)lds_dst,         // LDS dst (AS3)
      0, 0);                              // imm offset, imm cpol
#else
  *lds_dst = *gsrc;   // fallback: plain global load + ds_store
#endif
}

__device__ __forceinline__ void wait_async0() {
#if __has_builtin(__builtin_amdgcn_s_wait_asynccnt)
  __builtin_amdgcn_s_wait_asynccnt(0);
#else
  asm volatile("s_wait_asynccnt 0" ::: "memory");
#endif
}

__device__ __forceinline__ unsigned ballot32(bool p) {
#if __has_builtin(__builtin_amdgcn_ballot_w32)
  return __builtin_amdgcn_ballot_w32(p);
#else
  return (unsigned)__ballot(p);
#endif
}

// ---------------- small vec3 ----------------
struct V3 { float x, y, z; };
__device__ __forceinline__ V3 vsub(V3 a, V3 b) { return {a.x - b.x, a.y - b.y, a.z - b.z}; }
__device__ __forceinline__ V3 vcross(V3 a, V3 b) {
  return {a.y * b.z - a.z * b.y, a.z * b.x - a.x * b.z, a.x * b.y - a.y * b.x};
}
__device__ __forceinline__ float vdot(V3 a, V3 b) { return a.x * b.x + a.y * b.y + a.z * b.z; }

// Moller-Trumbore segment(p0,d) vs triangle(v0,e1,e2); matches reference EPS handling
__device__ __forceinline__ bool moller(V3 p0, V3 d, V3 v0, V3 e1, V3 e2,
                                       float& u, float& v, float& t) {
  V3 h = vcross(d, e2);
  float a = vdot(e1, h);
  bool ok = fabsf(a) > EPS_T;
  float f = 1.0f / (ok ? a : 1.0f);
  V3 s = vsub(p0, v0);
  u = f * vdot(s, h);
  V3 q = vcross(s, e1);
  v = f * vdot(d, q);
  t = f * vdot(e2, q);
  return ok & (u >= 0.0f) & (v >= 0.0f) & ((u + v) <= 1.0f) & (t >= 0.0f) & (t <= 1.0f);
}

// One wave (32 lanes) per query triangle; lanes sweep targets in ascending
// 32-wide groups; ballot + prefix popcount gives top_k-ordered compaction.
__global__ void __launch_bounds__(BLOCK_THREADS)
mesh_mesh_isect_kernel(const float* __restrict__ qtri, const float* __restrict__ ttri,
                       int* __restrict__ faces, float* __restrict__ bcs,
                       int Fq, int Ft) {
  __shared__ __align__(16) float ldsT[2][CHUNK * 9];

  const int tid  = threadIdx.x;
  const int lane = tid & 31;
  const int wid  = tid >> 5;
  const int q    = blockIdx.x * WAVES_PER_BLOCK + wid;
  const bool active = (q < Fq);

  V3 qv[3] = {{0.f,0.f,0.f},{0.f,0.f,0.f},{0.f,0.f,0.f}};
  if (active) {
    const float* qp = qtri + (size_t)q * 9;
    qv[0] = {qp[0], qp[1], qp[2]};
    qv[1] = {qp[3], qp[4], qp[5]};
    qv[2] = {qp[6], qp[7], qp[8]};
  }
  const V3 qe1 = vsub(qv[1], qv[0]);
  const V3 qe2 = vsub(qv[2], qv[0]);

  auto load_chunk = [&](int c, int buf) {
    const int base = c * CHUNK;
    const int ntri = (Ft - base < CHUNK) ? (Ft - base) : CHUNK;
    if (ntri == CHUNK) {
      const float4* g = (const float4*)(ttri + (size_t)base * 9);
      float4* l = (float4*)ldsT[buf];
      #pragma unroll
      for (int i = tid; i < (CHUNK * 9) / 4; i += BLOCK_THREADS)
        async_copy_b128(&l[i], &g[i]);
    } else if (ntri > 0) {
      const float* g = ttri + (size_t)base * 9;
      for (int i = tid; i < ntri * 9; i += BLOCK_THREADS)
        ldsT[buf][i] = g[i];
    }
  };

  const int nch = (Ft + CHUNK - 1) / CHUNK;
  load_chunk(0, 0);
  wait_async0();
  __syncthreads();

  int cnt = 0;                       // wave-uniform: hits recorded so far
  const int qbase = q * MAXC;

  for (int c = 0; c < nch; ++c) {
    const int buf = c & 1;
    if (c + 1 < nch) load_chunk(c + 1, buf ^ 1);   // async prefetch next chunk

    const int chunkBase = c * CHUNK;
    const int ntri = (Ft - chunkBase < CHUNK) ? (Ft - chunkBase) : CHUNK;
    const int ngroups = (ntri + 31) >> 5;

    for (int g = 0; g < ngroups; ++g) {
      const int tl = (g << 5) + lane;       // local target idx (ascending per group)
      const int t  = chunkBase + tl;
      bool hit = false;
      float b0 = 0.f, b1 = 0.f, b2 = 0.f;

      if (active && tl < ntri) {
        const float* tp = &ldsT[buf][tl * 9];   // stride 9: bank-conflict free
        V3 tv[3];
        tv[0] = {tp[0], tp[1], tp[2]};
        tv[1] = {tp[3], tp[4], tp[5]};
        tv[2] = {tp[6], tp[7], tp[8]};
        const V3 te1 = vsub(tv[1], tv[0]);
        const V3 te2 = vsub(tv[2], tv[0]);
        bool got = false;

        // pass 1: target edges vs query face -> bc = (1-u-v, u, v)
        #pragma unroll
        for (int k = 0; k < 3; ++k) {
          V3 p0 = tv[k];
          V3 d  = vsub(tv[(k + 1) % 3], p0);
          float u, v, tt;
          bool h = moller(p0, d, qv[0], qe1, qe2, u, v, tt);
          if (h && !got) { b0 = 1.0f - u - v; b1 = u; b2 = v; got = true; }
          hit |= h;
        }
        // pass 2: query edges vs target face -> bc = (1-t)*e_k + t*e_{k+1}
        #pragma unroll
        for (int k = 0; k < 3; ++k) {
          V3 p0 = qv[k];
          V3 d  = vsub(qv[(k + 1) % 3], p0);
          float u, v, tt;
          bool h = moller(p0, d, tv[0], te1, te2, u, v, tt);
          if (h && !got) {
            float e[3] = {0.f, 0.f, 0.f};
            e[k]           += 1.0f - tt;
            e[(k + 1) % 3] += tt;
            b0 = e[0]; b1 = e[1]; b2 = e[2];
            got = true;
          }
          hit |= h;
        }
      }

      // ordered wave compaction: first-32 ascending hits == stable top_k
      const unsigned mask = ballot32(hit);
      if (hit) {
        const int slot = cnt + __builtin_popcount(mask & ((1u << lane) - 1u));
        if (slot < MAXC) {
          faces[qbase + slot] = t;
          float* bp = bcs + (size_t)(qbase + slot) * 3;
          bp[0] = b0; bp[1] = b1; bp[2] = b2;
        }
      }
      cnt += __builtin_popcount(mask);
    }

    wait_async0();      // next chunk's async loads have landed in LDS
    __syncthreads();    // and everyone is done reading the current chunk
  }

  // pad unused slots: -1 faces, zero bcs (d_out is poisoned, must overwrite all)
  if (active) {
    for (int s = cnt + lane; s < MAXC; s += 32) {
      faces[qbase + s] = -1;
      float* bp = bcs + (size_t)(qbase + s) * 3;
      bp[0] = 0.f; bp[1] = 0.f; bp[2] = 0.f;
    }
  }
}

extern "C" void kernel_launch(void* const* d_in, const int* in_sizes, int n_in,
                              void* d_out, int out_size, void* d_ws, size_t ws_size,
                              hipStream_t stream) {
  const float* qtri = (const float*)d_in[0];   // [B*Fq, 3, 3] f32
  const float* ttri = (const float*)d_in[1];   // [B*Ft, 3, 3] f32
  const int Fq = in_sizes[0] / 9;              // B==1
  const int Ft = in_sizes[1] / 9;

  int*   faces = (int*)d_out;                       // [Fq, 32] i32 (bit-stored)
  float* bcs   = (float*)d_out + (size_t)Fq * MAXC; // [Fq, 32, 3] f32

  const int grid = (Fq + WAVES_PER_BLOCK - 1) / WAVES_PER_BLOCK;
  mesh_mesh_isect_kernel<<<grid, BLOCK_THREADS, 0, stream>>>(qtri, ttri, faces, bcs, Fq, Ft);
}